// Final_GCN_50757923504707
// MI455X (gfx1250) — compile-verified
//
#include <hip/hip_runtime.h>

#define HIDC 128
#define INC  11

typedef float v2f __attribute__((ext_vector_type(2)));
typedef float v8f __attribute__((ext_vector_type(8)));
typedef unsigned int v4u __attribute__((ext_vector_type(4)));
typedef int v4i __attribute__((ext_vector_type(4)));
typedef int v8i __attribute__((ext_vector_type(8)));

#ifndef USE_TDM
#define USE_TDM 1
#endif
#if USE_TDM && __has_builtin(__builtin_amdgcn_tensor_load_to_lds) && \
    __has_builtin(__builtin_amdgcn_s_wait_tensorcnt)
#define HAVE_TDM 1
#else
#define HAVE_TDM 0
#endif

// ---------------- degree / normalization ----------------
__global__ void k_init_deg(float* deg, int n) {
    int i = blockIdx.x * blockDim.x + threadIdx.x;
    if (i < n) deg[i] = 1.0f;              // self loop contributes 1
}

__global__ void k_count_deg(const int* __restrict__ col, float* deg, int E) {
    int e = blockIdx.x * blockDim.x + threadIdx.x;
    if (e < E) atomicAdd(&deg[col[e]], 1.0f);
}

__global__ void k_dinv(float* deg, int n) {
    int i = blockIdx.x * blockDim.x + threadIdx.x;
    if (i < n) { float d = deg[i]; deg[i] = d > 0.f ? rsqrtf(d) : 0.f; }
}

// ---------------- layer-1 linear (K=11, no WMMA payoff) ----------------
__global__ __launch_bounds__(HIDC)
void k_lin1(const float* __restrict__ x, const float* __restrict__ W1,
            float* __restrict__ y, int n) {
    int node = blockIdx.x;
    int c = threadIdx.x;
    __shared__ float sx[INC];
    if (c < INC) sx[c] = x[(size_t)node * INC + c];
    __syncthreads();
    float acc = 0.f;
#pragma unroll
    for (int k = 0; k < INC; ++k) acc += sx[k] * W1[k * HIDC + c];
    y[(size_t)node * HIDC + c] = acc;
}

// ---------------- message passing ----------------
__global__ void k_aggr_init(const float* __restrict__ y, const float* __restrict__ dinv,
                            float* __restrict__ aggr, long long total) {
    long long i = (long long)blockIdx.x * blockDim.x + threadIdx.x;
    if (i >= total) return;
    int node = (int)(i >> 7);
    float d = dinv[node];
    aggr[i] = d * d * y[i];
}

__global__ void k_edge_aggr(const int* __restrict__ row, const int* __restrict__ colp,
                            const float* __restrict__ dinv,
                            const float* __restrict__ y, float* __restrict__ aggr, int E) {
    long long tid = (long long)blockIdx.x * blockDim.x + threadIdx.x;
    int e = (int)(tid >> 5);
    int q = (int)(tid & 31);
    if (e >= E) return;
    int r = row[e], c = colp[e];
    float w = dinv[r] * dinv[c];
    const float4 v = *(const float4*)(y + (size_t)r * HIDC + q * 4);
    float* dst = aggr + (size_t)c * HIDC + q * 4;
    atomicAdd(dst + 0, w * v.x);
    atomicAdd(dst + 1, w * v.y);
    atomicAdd(dst + 2, w * v.z);
    atomicAdd(dst + 3, w * v.w);
}

__global__ void k_bias_act(float* __restrict__ h, const float* __restrict__ b,
                           long long total, int do_relu) {
    long long i = (long long)blockIdx.x * blockDim.x + threadIdx.x;
    if (i >= total) return;
    int c = (int)(i & (HIDC - 1));
    float v = h[i] + b[c];
    if (do_relu) v = fmaxf(v, 0.f);
    h[i] = v;
}

// ---------------- TDM staging of W (128x128 f32 -> LDS) ----------------
#if HAVE_TDM
__device__ __forceinline__ void tdm_stage_w(const float* W, float* sW) {
    unsigned long long ga = (unsigned long long)(uintptr_t)W;
    unsigned int lds = (unsigned int)(uintptr_t)sW;  // low 32 bits = LDS byte offset

    // D# group 0: count=1 | lds_addr | global_addr[56:0] | type=2
    v4u g0;
    g0[0] = 1u;
    g0[1] = lds;
    g0[2] = (unsigned int)(ga & 0xffffffffull);
    g0[3] = (unsigned int)((ga >> 32) & 0x01ffffffull) | (0x2u << 30);

    // D# group 1: data_size=4B | tensor_dim0/1=128 | tile_dim0/1=128 | dim0_stride=128
    v8i g1 = {};
    g1[0] = 0x00020000;         // data_size code 2 (4 bytes) in [17:16]
    g1[1] = (int)(128u << 16);  // tensor_dim0[15:0] in bits [63:48]
    g1[2] = (int)(128u << 16);  // tensor_dim1[15:0] in bits [95:80]
    g1[3] = (int)(128u << 16);  // tile_dim0=128 in bits [127:112]
    g1[4] = 128;                // tile_dim1=128 in bits [143:128]
    g1[5] = 128;                // tensor_dim0_stride low32 (bits [191:160])
    g1[6] = 0;
    g1[7] = 0;

    v4i z4 = {};
#if __clang_major__ >= 23
    v8i z8 = {};
    __builtin_amdgcn_tensor_load_to_lds(g0, g1, z4, z4, z8, 0);
#else
    __builtin_amdgcn_tensor_load_to_lds(g0, g1, z4, z4, 0);
#endif
    __builtin_amdgcn_s_wait_tensorcnt(0);
}
#endif

// ---------------- fp32 WMMA GEMM: Y[M,128] = A[M,128] @ W[128,128] ----------------
// 256 threads = 8 wave32. Wave w owns column tile n0 = 16*w and keeps its 32 B
// fragments (K swept in steps of 4) resident in 64 VGPRs. Each block sweeps 8
// row-tiles (128 rows); inner loop is pure global_load_b64 + v_wmma_f32_16x16x4_f32.
__global__ __launch_bounds__(256)
void k_gemm128_wmma(const float* __restrict__ A, const float* __restrict__ W,
                    const float* __restrict__ bias, float* __restrict__ Y,
                    int M, int do_relu) {
    __shared__ float sW[HIDC * HIDC];      // 64 KB of the 320 KB/WGP LDS

#if HAVE_TDM
    if (threadIdx.x < 32) {                // wave 0 issues the tensor DMA
        tdm_stage_w(W, sW);
    }
    __syncthreads();
#else
    for (int i = threadIdx.x; i < HIDC * HIDC; i += 256) sW[i] = W[i];
    __syncthreads();
#endif

    const int wave = threadIdx.x >> 5;     // wave32: wave id = column tile
    const int lane = threadIdx.x & 31;
    const int half = lane >> 4;            // 0: K pair {0,1}, 1: K pair {2,3}
    const int l    = lane & 15;
    const int n    = wave * 16 + l;        // this lane's output column

    // Pull this wave's 32 B fragments into registers (one-time DS cost)
    v2f bf[32];
#pragma unroll
    for (int kg = 0; kg < 32; ++kg) {
        const int kb = kg * 4 + half * 2;
        bf[kg].x = sW[(kb + 0) * HIDC + n];
        bf[kg].y = sW[(kb + 1) * HIDC + n];
    }

    const int rowBase = blockIdx.x * 128;
    const float bv = bias ? bias[n] : 0.f;

    for (int rt = 0; rt < 8; ++rt) {
        const int row0 = rowBase + rt * 16;
        int arow = row0 + l;
        if (arow >= M) arow = M - 1;       // clamp: EXEC stays all-ones for WMMA
        const float* aptr = A + (size_t)arow * HIDC + half * 2;

        v8f acc = {};
#pragma unroll
        for (int kg = 0; kg < 32; ++kg) {
            v2f a = *(const v2f*)(aptr + kg * 4);   // 8B aligned
            acc = __builtin_amdgcn_wmma_f32_16x16x4_f32(
                false, a, false, bf[kg], (short)0, acc, false, false);
        }

        // C/D layout: lane<16 -> rows row0+0..7 (VGPR v = M=v); lane>=16 -> +8
#pragma unroll
        for (int v = 0; v < 8; ++v) {
            const int r = row0 + half * 8 + v;
            if (r < M) {
                float val = acc[v] + bv;
                if (do_relu) val = fmaxf(val, 0.f);
                Y[(size_t)r * HIDC + n] = val;
            }
        }
    }
}

// ---------------- pooling + head ----------------
__global__ void k_zero(float* p, int n) {
    int i = blockIdx.x * blockDim.x + threadIdx.x;
    if (i < n) p[i] = 0.f;
}

__global__ void k_pool_sum(const float* __restrict__ h, const int* __restrict__ batch,
                           float* __restrict__ pooled, long long total) {
    long long i = (long long)blockIdx.x * blockDim.x + threadIdx.x;
    if (i >= total) return;
    int node = (int)(i >> 7);
    int c = (int)(i & (HIDC - 1));
    int g = batch[node];
    atomicAdd(&pooled[g * HIDC + c], h[i]);
}

__global__ void k_pool_cnt(const int* __restrict__ batch, float* cnt, int n) {
    int i = blockIdx.x * blockDim.x + threadIdx.x;
    if (i < n) atomicAdd(&cnt[batch[i]], 1.f);
}

__global__ void k_pool_div(float* pooled, const float* __restrict__ cnt, int total) {
    int i = blockIdx.x * blockDim.x + threadIdx.x;
    if (i < total) {
        int g = i >> 7;
        pooled[i] /= fmaxf(cnt[g], 1.f);
    }
}

__global__ __launch_bounds__(HIDC)
void k_head_out(const float* __restrict__ g, const float* __restrict__ Wo,
                const float* __restrict__ bo, float* __restrict__ out) {
    int i = threadIdx.x;                   // one block: 128 graphs
    float acc = 0.f;
    for (int k = 0; k < HIDC; ++k) acc += g[i * HIDC + k] * Wo[k];
    out[i] = acc + bo[0];
}

// ---------------- launch ----------------
extern "C" void kernel_launch(void* const* d_in, const int* in_sizes, int n_in,
                              void* d_out, int out_size, void* d_ws, size_t ws_size,
                              hipStream_t stream) {
    const float* x     = (const float*)d_in[0];
    const int*   ei    = (const int*)d_in[1];   // [2, E]
    const int*   batch = (const int*)d_in[2];
    const float* W1 = (const float*)d_in[3];
    const float* b1 = (const float*)d_in[4];
    const float* W2 = (const float*)d_in[5];
    const float* b2 = (const float*)d_in[6];
    const float* W4 = (const float*)d_in[7];
    const float* b4 = (const float*)d_in[8];
    const float* Wd1 = (const float*)d_in[9];
    const float* bd1 = (const float*)d_in[10];
    const float* Wd2 = (const float*)d_in[11];
    const float* bd2 = (const float*)d_in[12];
    const float* Wo  = (const float*)d_in[13];
    const float* bo  = (const float*)d_in[14];
    float* out = (float*)d_out;

    const int N = in_sizes[0] / INC;
    const int E = in_sizes[1] / 2;
    const int G = out_size;                     // 128 graphs

    char* ws = (char*)d_ws;
    size_t off = 0;
    auto take = [&](size_t bytes) -> char* {
        char* p = ws + off;
        off = (off + bytes + 255) & ~(size_t)255;
        return p;
    };
    float* buf0   = (float*)take((size_t)N * HIDC * sizeof(float));
    float* buf1   = (float*)take((size_t)N * HIDC * sizeof(float));
    float* dinv   = (float*)take((size_t)N * sizeof(float));
    float* pooled = (float*)take((size_t)G * HIDC * sizeof(float));
    float* cnt    = (float*)take((size_t)G * sizeof(float));
    float* g1     = (float*)take((size_t)G * HIDC * sizeof(float));
    float* g2     = (float*)take((size_t)G * HIDC * sizeof(float));
    (void)n_in; (void)ws_size;

    const int* row = ei;
    const int* col = ei + E;

    const long long totNH = (long long)N * HIDC;
    const int blkNH  = (int)((totNH + 255) / 256);
    const int blkE32 = (int)(((long long)E * 32 + 255) / 256);

    // symmetric GCN normalization: dinv = rsqrt(deg + selfloop)
    k_init_deg<<<(N + 255) / 256, 256, 0, stream>>>(dinv, N);
    k_count_deg<<<(E + 255) / 256, 256, 0, stream>>>(col, dinv, E);
    k_dinv<<<(N + 255) / 256, 256, 0, stream>>>(dinv, N);

    // layer 1: lin(x,W1) -> buf0; aggregate -> buf1; +b1, relu
    k_lin1<<<N, HIDC, 0, stream>>>(x, W1, buf0, N);
    k_aggr_init<<<blkNH, 256, 0, stream>>>(buf0, dinv, buf1, totNH);
    k_edge_aggr<<<blkE32, 256, 0, stream>>>(row, col, dinv, buf0, buf1, E);
    k_bias_act<<<blkNH, 256, 0, stream>>>(buf1, b1, totNH, 1);

    // layer 2: WMMA gemm -> buf0; aggregate -> buf1; +b2, relu
    k_gemm128_wmma<<<(N + 127) / 128, 256, 0, stream>>>(buf1, W2, nullptr, buf0, N, 0);
    k_aggr_init<<<blkNH, 256, 0, stream>>>(buf0, dinv, buf1, totNH);
    k_edge_aggr<<<blkE32, 256, 0, stream>>>(row, col, dinv, buf0, buf1, E);
    k_bias_act<<<blkNH, 256, 0, stream>>>(buf1, b2, totNH, 1);

    // layer 3: WMMA gemm -> buf0; aggregate -> buf1; +b4 (no relu)
    k_gemm128_wmma<<<(N + 127) / 128, 256, 0, stream>>>(buf1, W4, nullptr, buf0, N, 0);
    k_aggr_init<<<blkNH, 256, 0, stream>>>(buf0, dinv, buf1, totNH);
    k_edge_aggr<<<blkE32, 256, 0, stream>>>(row, col, dinv, buf0, buf1, E);
    k_bias_act<<<blkNH, 256, 0, stream>>>(buf1, b4, totNH, 0);

    // global mean pool
    k_zero<<<(G * HIDC + 255) / 256, 256, 0, stream>>>(pooled, G * HIDC);
    k_zero<<<1, 256, 0, stream>>>(cnt, G);
    k_pool_sum<<<blkNH, 256, 0, stream>>>(buf1, batch, pooled, totNH);
    k_pool_cnt<<<(N + 255) / 256, 256, 0, stream>>>(batch, cnt, N);
    k_pool_div<<<(G * HIDC + 255) / 256, 256, 0, stream>>>(pooled, cnt, G * HIDC);

    // dense head (128x128 GEMMs via WMMA, bias+relu fused), then [128,1] projection
    k_gemm128_wmma<<<(G + 127) / 128, 256, 0, stream>>>(pooled, Wd1, bd1, g1, G, 1);
    k_gemm128_wmma<<<(G + 127) / 128, 256, 0, stream>>>(g1, Wd2, bd2, g2, G, 1);
    k_head_out<<<1, HIDC, 0, stream>>>(g2, Wo, bo, out);
}